// S4SSM_38646115730086
// MI455X (gfx1250) — compile-verified
//
#include <hip/hip_runtime.h>
#include <hip/hip_bf16.h>

#define HCH 511
#define LSEQ 2048
#define NMODES 32
#define NLAYERS 4
#define M2 1022            // 2*H
#define KD 511             // inner dim of GLU matmul
#define KPAD 512
#define LN_EPS 1e-5f

typedef __attribute__((ext_vector_type(2))) float v2f;
typedef __attribute__((ext_vector_type(8))) float v8f;

__device__ __forceinline__ float gelu_tanh(float x) {
    // jax.nn.gelu default (approximate=True)
    float x3 = x * x * x;
    float t = tanhf(0.7978845608028654f * (x + 0.044715f * x3));
    return 0.5f * x * (1.0f + t);
}

__device__ __forceinline__ float sigmoidf(float x) {
    return 1.0f / (1.0f + expf(-x));
}

// ---------------------------------------------------------------------------
// Precompute per-(h,n) discrete coefficients: w = exp(dt*A), c = 2*C*(w-1)/A
// ---------------------------------------------------------------------------
__global__ void s4_setup(const float* __restrict__ log_dt,
                         const float* __restrict__ Arl,
                         const float* __restrict__ Aim,
                         const float* __restrict__ Cre,
                         const float* __restrict__ Cim,
                         float* __restrict__ w_re, float* __restrict__ w_im,
                         float* __restrict__ c_re, float* __restrict__ c_im) {
    int idx = blockIdx.x * blockDim.x + threadIdx.x;
    if (idx >= HCH * NMODES) return;
    int h = idx / NMODES;
    float a_re = -expf(Arl[idx]);
    float a_im = Aim[idx];
    float dt = expf(log_dt[h]);
    float dre = a_re * dt, dim = a_im * dt;
    float er = expf(dre);
    float sn, cs;
    sincosf(dim, &sn, &cs);
    float wre = er * cs, wim = er * sn;
    // (w - 1) / A  via conj
    float nre = wre - 1.0f, nim = wim;
    float inv = 1.0f / (a_re * a_re + a_im * a_im);
    float tre = (nre * a_re + nim * a_im) * inv;
    float tim = (nim * a_re - nre * a_im) * inv;
    float Cr = Cre[idx], Ci = Cim[idx];
    w_re[idx] = wre;
    w_im[idx] = wim;
    c_re[idx] = 2.0f * (Cr * tre - Ci * tim);
    c_im[idx] = 2.0f * (Cr * tim + Ci * tre);
}

// ---------------------------------------------------------------------------
// Diagonal SSM scan: one wave per channel, lane = mode (N == 32 == wave32).
// y[h,l] = gelu( SSM(z)[h,l] + D_skip[h]*z[h,l] )
// ---------------------------------------------------------------------------
__global__ __launch_bounds__(256) void s4_scan(
        const float* __restrict__ z,
        const float* __restrict__ w_re, const float* __restrict__ w_im,
        const float* __restrict__ c_re, const float* __restrict__ c_im,
        const float* __restrict__ D_skip, float* __restrict__ y) {
    int warp = threadIdx.x >> 5;
    int lane = threadIdx.x & 31;
    int h = blockIdx.x * 8 + warp;
    if (h >= HCH) return;
    int cidx = h * NMODES + lane;
    float wr = w_re[cidx], wi = w_im[cidx];
    float cr = c_re[cidx], ci = c_im[cidx];
    float d = D_skip[h];
    float xr = 0.0f, xi = 0.0f;
    const float* zrow = z + (size_t)h * LSEQ;
    float* yrow = y + (size_t)h * LSEQ;
    for (int l0 = 0; l0 < LSEQ; l0 += 32) {
        float zv = zrow[l0 + lane];
        float out = 0.0f;
#pragma unroll
        for (int t = 0; t < 32; ++t) {
            float zt = __shfl(zv, t, 32);
            float nxr = fmaf(wr, xr, fmaf(-wi, xi, zt));
            float nxi = fmaf(wr, xi, wi * xr);
            xr = nxr; xi = nxi;
            float contrib = fmaf(cr, xr, -ci * xi);
#pragma unroll
            for (int off = 16; off > 0; off >>= 1)
                contrib += __shfl_xor(contrib, off, 32);
            if (lane == t) out = contrib;
        }
        yrow[l0 + lane] = gelu_tanh(fmaf(d, zv, out));
    }
}

// ---------------------------------------------------------------------------
// GLU GEMM: Y2[1022 x 2048] = glu_w[1022 x 511] * Y[511 x 2048] + glu_b
// fp32 WMMA 16x16x4, 128x128 block tile, 8 waves (2x4), 64x32 (m x n) per wave.
// ---------------------------------------------------------------------------
#define BM 128
#define BN 128
#define BK 16
#define LDP 18   // padded LDS row (floats)

__global__ __launch_bounds__(256) void s4_glu_gemm(
        const float* __restrict__ Wg, const float* __restrict__ bias,
        const float* __restrict__ Y, float* __restrict__ Y2) {
    __shared__ float Ws[BM][LDP];   // [m][k]
    __shared__ float Ys[BN][LDP];   // [n][k]  (transposed stage)
    int tid = threadIdx.x;
    int lane = tid & 31;
    int wave = tid >> 5;
    int wave_m = wave >> 2;               // 0..1
    int wave_n = wave & 3;                // 0..3
    int lrow = lane & 15;
    int khalf = (lane >> 4) << 1;         // 0 or 2
    int madd = (lane >> 4) << 3;          // D-frag: m = i + 8*(lane>=16)
    int blockM = blockIdx.x * BM;
    int blockN = blockIdx.y * BN;

    v8f acc[4][2];
#pragma unroll
    for (int mt = 0; mt < 4; ++mt) {
#pragma unroll
        for (int i = 0; i < 8; ++i) {
            int gm = blockM + wave_m * 64 + mt * 16 + i + madd;
            float bv = (gm < M2) ? bias[gm] : 0.0f;
#pragma unroll
            for (int nt = 0; nt < 2; ++nt) acc[mt][nt][i] = bv;
        }
    }

    for (int kb = 0; kb < KPAD; kb += BK) {
        // stage W tile [BM x BK]: thread -> (m = tid/2, 8 consecutive k)
        {
            int m = tid >> 1;
            int ks = (tid & 1) * 8;
            int gm = blockM + m;
#pragma unroll
            for (int j = 0; j < 8; ++j) {
                int gk = kb + ks + j;
                Ws[m][ks + j] = (gm < M2 && gk < KD) ? Wg[gm * KD + gk] : 0.0f;
            }
        }
        // stage Y tile [BK x BN] transposed: thread -> (k = tid/16, 8 consecutive n)
        {
            int k = tid >> 4;
            int ns = (tid & 15) * 8;
            int gk = kb + k;
            if (gk < KD) {
                const float* src = Y + (size_t)gk * LSEQ + blockN + ns;
#pragma unroll
                for (int j = 0; j < 8; ++j) Ys[ns + j][k] = src[j];
            } else {
#pragma unroll
                for (int j = 0; j < 8; ++j) Ys[ns + j][k] = 0.0f;
            }
        }
        __syncthreads();
#pragma unroll
        for (int kc = 0; kc < 4; ++kc) {
            int kk = kc * 4 + khalf;
            v2f a[4], b[2];
#pragma unroll
            for (int mt = 0; mt < 4; ++mt)
                a[mt] = *(const v2f*)&Ws[wave_m * 64 + mt * 16 + lrow][kk];
#pragma unroll
            for (int nt = 0; nt < 2; ++nt)
                b[nt] = *(const v2f*)&Ys[wave_n * 32 + nt * 16 + lrow][kk];
#pragma unroll
            for (int mt = 0; mt < 4; ++mt)
#pragma unroll
                for (int nt = 0; nt < 2; ++nt)
                    acc[mt][nt] = __builtin_amdgcn_wmma_f32_16x16x4_f32(
                        false, a[mt], false, b[nt],
                        (short)0, acc[mt][nt], false, false);
        }
        __syncthreads();
    }

    int gn_base = blockN + wave_n * 32;
#pragma unroll
    for (int mt = 0; mt < 4; ++mt) {
#pragma unroll
        for (int i = 0; i < 8; ++i) {
            int gm = blockM + wave_m * 64 + mt * 16 + i + madd;
            if (gm < M2) {
#pragma unroll
                for (int nt = 0; nt < 2; ++nt) {
                    int gn = gn_base + nt * 16 + lrow;
                    Y2[(size_t)gm * LSEQ + gn] = acc[mt][nt][i];
                }
            }
        }
    }
}

// ---------------------------------------------------------------------------
// GLU + residual + channel LayerNorm, one column l per block (in-place on Z)
// ---------------------------------------------------------------------------
__global__ __launch_bounds__(256) void s4_glu_ln(
        const float* __restrict__ Y2, float* __restrict__ Z,
        const float* __restrict__ g, const float* __restrict__ b) {
    __shared__ float red[256];
    int l = blockIdx.x;
    int tid = threadIdx.x;
    int h1 = tid + 256;

    float v0, v1 = 0.0f;
    {
        float p = Y2[(size_t)tid * LSEQ + l];
        float q = Y2[(size_t)(tid + HCH) * LSEQ + l];
        v0 = p * sigmoidf(q) + Z[(size_t)tid * LSEQ + l];
    }
    if (h1 < HCH) {
        float p = Y2[(size_t)h1 * LSEQ + l];
        float q = Y2[(size_t)(h1 + HCH) * LSEQ + l];
        v1 = p * sigmoidf(q) + Z[(size_t)h1 * LSEQ + l];
    }

    // mean
    red[tid] = v0 + v1;
    __syncthreads();
    for (int s = 128; s > 0; s >>= 1) {
        if (tid < s) red[tid] += red[tid + s];
        __syncthreads();
    }
    float mu = red[0] * (1.0f / (float)HCH);
    __syncthreads();

    // variance (two-pass, matches reference mean((x-mu)^2))
    float d0 = v0 - mu;
    float d1 = (h1 < HCH) ? (v1 - mu) : 0.0f;
    red[tid] = d0 * d0 + d1 * d1;
    __syncthreads();
    for (int s = 128; s > 0; s >>= 1) {
        if (tid < s) red[tid] += red[tid + s];
        __syncthreads();
    }
    float rstd = rsqrtf(red[0] * (1.0f / (float)HCH) + LN_EPS);

    Z[(size_t)tid * LSEQ + l] = d0 * rstd * g[tid] + b[tid];
    if (h1 < HCH)
        Z[(size_t)h1 * LSEQ + l] = d1 * rstd * g[h1] + b[h1];
}

// ---------------------------------------------------------------------------
extern "C" void kernel_launch(void* const* d_in, const int* in_sizes, int n_in,
                              void* d_out, int out_size, void* d_ws, size_t ws_size,
                              hipStream_t stream) {
    (void)in_sizes; (void)n_in; (void)out_size; (void)ws_size;
    const float* Zin    = (const float*)d_in[0];
    const float* log_dt = (const float*)d_in[1];
    const float* Arl    = (const float*)d_in[2];
    const float* Aim    = (const float*)d_in[3];
    const float* Cre    = (const float*)d_in[4];
    const float* Cim    = (const float*)d_in[5];
    const float* Dsk    = (const float*)d_in[6];
    const float* glu_w  = (const float*)d_in[7];
    const float* glu_b  = (const float*)d_in[8];
    const float* ln_g   = (const float*)d_in[9];
    const float* ln_b   = (const float*)d_in[10];

    float* ws = (float*)d_ws;
    const size_t HN = (size_t)HCH * NMODES;
    float* w_re  = ws;
    float* w_im  = ws + HN;
    float* c_re  = ws + 2 * HN;
    float* c_im  = ws + 3 * HN;
    float* zbuf  = ws + 4 * HN;
    float* ybuf  = zbuf + (size_t)HCH * LSEQ;
    float* y2buf = ybuf + (size_t)HCH * LSEQ;

    s4_setup<<<(int)((HN + 255) / 256), 256, 0, stream>>>(
        log_dt, Arl, Aim, Cre, Cim, w_re, w_im, c_re, c_im);

    hipMemcpyAsync(zbuf, Zin, sizeof(float) * (size_t)HCH * LSEQ,
                   hipMemcpyDeviceToDevice, stream);

    for (int layer = 0; layer < NLAYERS; ++layer) {
        s4_scan<<<64, 256, 0, stream>>>(zbuf, w_re, w_im, c_re, c_im, Dsk, ybuf);
        s4_glu_gemm<<<dim3(8, 16, 1), 256, 0, stream>>>(glu_w, glu_b, ybuf, y2buf);
        s4_glu_ln<<<LSEQ, 256, 0, stream>>>(y2buf, zbuf, ln_g, ln_b);
    }

    hipMemcpyAsync(d_out, zbuf, sizeof(float) * (size_t)HCH * LSEQ,
                   hipMemcpyDeviceToDevice, stream);
}